// PhysicsLoss_31095563223112
// MI455X (gfx1250) — compile-verified
//
#include <hip/hip_runtime.h>
#include <math.h>

// ---------------------------------------------------------------------------
// Physics-informed FiLM-DeepONet loss for MI455X (gfx1250, wave32, WMMA).
//
// All heavy math = per-layer GEMMs  Z(256 x 32cols) = W_l(256x256) * H(256x32)
// executed with V_WMMA_F32_16X16X4_F32 (exact f32 MACs, f32 accumulate) so the
// result tracks the f32 JAX reference; derivative (tangent) channels ride as
// extra GEMM columns. Activations live in LDS (col-major, stride 260 ->
// conflict-free ds_load_b64 B fragments), weights stream from L2 (1 MB).
// ---------------------------------------------------------------------------

typedef __attribute__((ext_vector_type(2))) float v2f;
typedef __attribute__((ext_vector_type(8))) float v8f;

#define HD      256      // hidden width
#define LNUM    4        // trunk layers
#define NCPAD   32       // padded GEMM columns per workgroup (2 N-tiles)
#define SSTR    260      // LDS column stride (260 % 64 == 4 -> no bank conflicts,
                         //  even & mult-of-4 -> b64/b128 alignment preserved)

__device__ __forceinline__ v8f wmma_f32(v2f a, v2f b, v8f c) {
  // D = A(16x4 f32) * B(4x16 f32) + C(16x16 f32)
  return __builtin_amdgcn_wmma_f32_16x16x4_f32(
      /*neg_a=*/false, a, /*neg_b=*/false, b,
      /*c_mod=*/(short)0, c, /*reuse_a=*/false, /*reuse_b=*/false);
}

// --------------------------- FiLM encoder ----------------------------------
// h  = tanh(enc_W1 @ g + b1)            (256x256 GEMV per batch)
// fb = enc_W2 @ h + b2  -> gamma,beta   (2048x256 GEMV per batch)
__global__ __launch_bounds__(256)
void film_encoder(const float* __restrict__ g_batch,
                  const float* __restrict__ enc_W1, const float* __restrict__ enc_b1,
                  const float* __restrict__ enc_W2, const float* __restrict__ enc_b2,
                  float* __restrict__ gb) {
  __shared__ float sg[HD];
  __shared__ float sh[HD];
  const int b = blockIdx.x, t = threadIdx.x;
  sg[t] = g_batch[b * 256 + t];           // g.reshape(-1), 2*m == 256
  __syncthreads();
  float a = enc_b1[t];
  for (int j = 0; j < HD; ++j) a += enc_W1[t * HD + j] * sg[j];
  sh[t] = tanhf(a);
  __syncthreads();
  for (int r = t; r < 2 * LNUM * HD; r += 256) {
    float acc = enc_b2[r];
    for (int j = 0; j < HD; ++j) acc += enc_W2[r * HD + j] * sh[j];
    gb[b * (2 * LNUM * HD) + r] = acc;    // [c][l][j] flat, c=0 gamma, c=1 beta
  }
}

// ------------------------------ trunk --------------------------------------
// NCH = 6 (value,dx,dy,dxx,dxy,dyy) interior Hessian pass, TP=4 points/WG
// NCH = 3 (value,dx,dy)             forcing Jacobian pass, TP=8 points/WG
template <int NCH, int TP, int BPB>
__global__ __launch_bounds__(256)
void trunk_kernel(const float* __restrict__ xy,
                  const float* __restrict__ W_in, const float* __restrict__ b_in,
                  const float* __restrict__ Ws, const float* __restrict__ bs,
                  const float* __restrict__ W_out,
                  const float* __restrict__ gb,
                  const float* __restrict__ g_batch,
                  float* __restrict__ partOut) {
  constexpr int REAL = NCH * TP;                  // 24 real columns
  constexpr int NQ = (NCH == 6) ? 6 * TP : 4 * TP;
  __shared__ float sH[NCPAD * SSTR];              // current activations (col-major)
  __shared__ float sZ[NCPAD * SSTR];              // GEMM output / reduction scratch

  const int tid = threadIdx.x;
  const int b = blockIdx.x / BPB;
  const int pt0 = (blockIdx.x % BPB) * TP;

  // ---- zero pad columns once (WMMA reads all 32 cols) ----
  for (int e = tid; e < (NCPAD - REAL) * HD; e += 256) {
    int c = REAL + e / HD, j = e % HD;
    sH[c * SSTR + j] = 0.f;
  }
  // ---- input layer: h = tanh(W_in@xy + b_in) + tangent chain ----
  for (int e = tid; e < TP * HD; e += 256) {
    int p = e >> 8, j = e & 255;
    float x = xy[(pt0 + p) * 2 + 0], y = xy[(pt0 + p) * 2 + 1];
    float wx = W_in[j * 2 + 0], wy = W_in[j * 2 + 1];
    float t = tanhf(wx * x + wy * y + b_in[j]);
    float d = 1.f - t * t;
    sH[(p * NCH + 0) * SSTR + j] = t;
    sH[(p * NCH + 1) * SSTR + j] = d * wx;
    sH[(p * NCH + 2) * SSTR + j] = d * wy;
    if constexpr (NCH == 6) {
      float c2 = -2.f * t * d;
      sH[(p * NCH + 3) * SSTR + j] = c2 * wx * wx;
      sH[(p * NCH + 4) * SSTR + j] = c2 * wx * wy;
      sH[(p * NCH + 5) * SSTR + j] = c2 * wy * wy;
    }
  }
  __syncthreads();

  const int lane = tid & 31;
  const int wave = tid >> 5;
  const int lm = lane & 15;    // M (A) / N (B,D) index within tile
  const int lh = lane >> 4;    // half selector (K pair for A/B, M+8 for D)
  const int m0 = wave * 32;    // each wave owns two 16-row M-tiles

  for (int l = 0; l < LNUM; ++l) {
    // ---- GEMM: sZ = Ws[l] * sH  (M=256, N=32, K=256 via 16x16x4 WMMA) ----
    const float* Wl = Ws + (size_t)l * HD * HD;
    const float* wr0 = Wl + (m0 + lm) * HD + lh * 2;        // A frag rows m0..m0+15
    const float* wr1 = wr0 + 16 * HD;                       // A frag rows +16
    const float* bp0 = &sH[lm * SSTR + lh * 2];             // B frag cols 0..15
    const float* bp1 = &sH[(16 + lm) * SSTR + lh * 2];      // B frag cols 16..31
    v8f a00 = {}, a01 = {}, a10 = {}, a11 = {};
    for (int k = 0; k < HD; k += 4) {
      v2f fa0 = *(const v2f*)(wr0 + k);   // W[m0+lm][k+2*lh .. +1]
      v2f fa1 = *(const v2f*)(wr1 + k);
      v2f fb0 = *(const v2f*)(bp0 + k);   // H[col][k+2*lh .. +1]
      v2f fb1 = *(const v2f*)(bp1 + k);
      a00 = wmma_f32(fa0, fb0, a00);
      a01 = wmma_f32(fa0, fb1, a01);
      a10 = wmma_f32(fa1, fb0, a10);
      a11 = wmma_f32(fa1, fb1, a11);
    }
    // D layout: VGPR v, lane -> (M = m0 + 8*lh + v, N = n0 + lm); 8 consecutive M
    {
      float* d00 = &sZ[lm * SSTR + m0 + 8 * lh];
      float* d01 = &sZ[(16 + lm) * SSTR + m0 + 8 * lh];
#pragma unroll
      for (int v = 0; v < 8; ++v) {
        d00[v] = a00[v];
        d00[16 + v] = a10[v];
        d01[v] = a01[v];
        d01[16 + v] = a11[v];
      }
    }
    __syncthreads();

    // ---- elementwise: FiLM scale + tanh forward-tangent chain (sZ -> sH) ----
    const float* gl = gb + b * (2 * LNUM * HD) + l * HD;  // gamma[l][*]
    const float* be = gl + LNUM * HD;                     // beta[l][*]
    const float* bl = bs + l * HD;
    for (int e = tid; e < TP * HD; e += 256) {
      int p = e >> 8, j = e & 255;
      float s = 1.f + gl[j];
      float zv = (sZ[(p * NCH + 0) * SSTR + j] + bl[j]) * s + be[j];
      float t = tanhf(zv);
      float d = 1.f - t * t;
      float zx = sZ[(p * NCH + 1) * SSTR + j] * s;
      float zy = sZ[(p * NCH + 2) * SSTR + j] * s;
      sH[(p * NCH + 0) * SSTR + j] = t;
      sH[(p * NCH + 1) * SSTR + j] = d * zx;
      sH[(p * NCH + 2) * SSTR + j] = d * zy;
      if constexpr (NCH == 6) {
        float zxx = sZ[(p * NCH + 3) * SSTR + j] * s;
        float zxy = sZ[(p * NCH + 4) * SSTR + j] * s;
        float zyy = sZ[(p * NCH + 5) * SSTR + j] * s;
        float c2 = -2.f * t * d;
        sH[(p * NCH + 3) * SSTR + j] = d * zxx + c2 * zx * zx;
        sH[(p * NCH + 4) * SSTR + j] = d * zxy + c2 * zx * zy;
        sH[(p * NCH + 5) * SSTR + j] = d * zyy + c2 * zy * zy;
      }
    }
    __syncthreads();
  }

  // ---- output head: W_out row-dots with derivative channels ----
  {
    const int j = tid;  // exactly 256 threads == hidden width
    float w0 = W_out[j], w1 = W_out[HD + j];
    float part[NQ];
#pragma unroll
    for (int p = 0; p < TP; ++p) {
      if constexpr (NCH == 6) {
        float hxx = sH[(p * 6 + 3) * SSTR + j];
        float hxy = sH[(p * 6 + 4) * SSTR + j];
        float hyy = sH[(p * 6 + 5) * SSTR + j];
        part[p * 6 + 0] = w0 * hxx; part[p * 6 + 1] = w0 * hxy; part[p * 6 + 2] = w0 * hyy;
        part[p * 6 + 3] = w1 * hxx; part[p * 6 + 4] = w1 * hxy; part[p * 6 + 5] = w1 * hyy;
      } else {
        float hx = sH[(p * 3 + 1) * SSTR + j];
        float hy = sH[(p * 3 + 2) * SSTR + j];
        part[p * 4 + 0] = w0 * hx; part[p * 4 + 1] = w0 * hy;
        part[p * 4 + 2] = w1 * hx; part[p * 4 + 3] = w1 * hy;
      }
    }
#pragma unroll
    for (int q = 0; q < NQ; ++q) sZ[q * HD + j] = part[q];  // NQ*256 <= sZ size
  }
  __syncthreads();
  if (tid < NQ) {            // deterministic fixed-order reduction per dot
    float s = 0.f;
    for (int i = 0; i < HD; ++i) s += sZ[tid * HD + i];
    sH[tid] = s;
  }
  __syncthreads();
  if (tid == 0) {
    const float PREF = 1.0f / 0.91f;   // E/(1-NU^2), E=1, NU=0.3
    const float NU = 0.3f;
    float acc = 0.f;
#pragma unroll
    for (int p = 0; p < TP; ++p) {
      if constexpr (NCH == 6) {
        float uxx = sH[p * 6 + 0], uxy = sH[p * 6 + 1], uyy = sH[p * 6 + 2];
        float vxx = sH[p * 6 + 3], vxy = sH[p * 6 + 4], vyy = sH[p * 6 + 5];
        float ru = PREF * (uxx + 0.5f * (1.f + NU) * vxy + 0.5f * (1.f - NU) * uyy);
        float rv = PREF * (vyy + 0.5f * (1.f + NU) * uxy + 0.5f * (1.f - NU) * vxx);
        acc += ru * ru + rv * rv;
      } else {
        float ux = sH[p * 4 + 0], uy = sH[p * 4 + 1];
        float vx = sH[p * 4 + 2], vy = sH[p * 4 + 3];
        int pm = pt0 + p;
        float gx = g_batch[b * 256 + pm];
        float gy = g_batch[b * 256 + 128 + pm];
        float bu = PREF * (ux + NU * vy) + gx;
        float bv = PREF * 0.5f * (1.f - NU) * (uy + vx) + gy;
        acc += bu * bu + bv * bv;
      }
    }
    partOut[blockIdx.x] = acc;   // per-block partial, summed deterministically later
  }
}

// ------------------------------ finalize -----------------------------------
__global__ void finalize_loss(const float* __restrict__ partInt,
                              const float* __restrict__ partForce,
                              float* __restrict__ out) {
  if (blockIdx.x == 0 && threadIdx.x == 0) {
    float si = 0.f;
    for (int i = 0; i < 1024; ++i) si += partInt[i];
    float sf = 0.f;
    for (int i = 0; i < 64; ++i) sf += partForce[i];
    // loss_int = sum(ru^2+rv^2)/(B*N_int); loss_forcing = sum(bu^2+bv^2)/(B*m)
    out[0] = si * (1.0f / 4096.0f) + sf * (1.0f / 512.0f);
  }
}

// ------------------------------ launch -------------------------------------
extern "C" void kernel_launch(void* const* d_in, const int* in_sizes, int n_in,
                              void* d_out, int out_size, void* d_ws, size_t ws_size,
                              hipStream_t stream) {
  (void)in_sizes; (void)n_in; (void)out_size; (void)ws_size;
  const float* g_batch = (const float*)d_in[0];
  const float* xy_int  = (const float*)d_in[1];
  const float* xy_forc = (const float*)d_in[2];
  const float* enc_W1  = (const float*)d_in[3];
  const float* enc_b1  = (const float*)d_in[4];
  const float* enc_W2  = (const float*)d_in[5];
  const float* enc_b2  = (const float*)d_in[6];
  const float* W_in    = (const float*)d_in[7];
  const float* b_in    = (const float*)d_in[8];
  const float* Ws      = (const float*)d_in[9];
  const float* bs      = (const float*)d_in[10];
  const float* W_out   = (const float*)d_in[11];
  // d_in[12] = b_out: drops out of all derivative terms.

  float* ws        = (float*)d_ws;
  float* gb        = ws;                 // [B][2][L][H] = 8192 floats
  float* partInt   = ws + 4 * 2048;      // 1024 floats (one per interior block)
  float* partForce = partInt + 1024;     // 64 floats

  film_encoder<<<4, 256, 0, stream>>>(g_batch, enc_W1, enc_b1, enc_W2, enc_b2, gb);
  // interior: B=4 * N_int=1024 points, 6 channels, 4 points/WG -> 1024 blocks
  trunk_kernel<6, 4, 256><<<1024, 256, 0, stream>>>(
      xy_int, W_in, b_in, Ws, bs, W_out, gb, g_batch, partInt);
  // forcing: B=4 * m=128 points, 3 channels, 8 points/WG -> 64 blocks
  trunk_kernel<3, 8, 16><<<64, 256, 0, stream>>>(
      xy_forc, W_in, b_in, Ws, bs, W_out, gb, g_batch, partForce);
  finalize_loss<<<1, 32, 0, stream>>>(partInt, partForce, (float*)d_out);
}